// LightGCN_4011499454850
// MI455X (gfx1250) — compile-verified
//
#include <hip/hip_runtime.h>
#include <stdint.h>

// LightGCN propagation for MI455X (gfx1250).
// d_ws requirement: 2 * N * 64 * 4 bytes = 153.6 MB (ping-pong embedding tables).
// acc lives directly in d_out (concat(users, items) == flat all-node layout).

#define N_USERS   200000
#define N_ITEMS   100000
#define NNZ_CNT   4000000
#define LDIM      64
#define NLAYERS   3
#define NTOTAL    (N_USERS + N_ITEMS)           // 300000
#define NELEM     ((size_t)NTOTAL * LDIM)       // 19,200,000 floats per table

typedef float        v2f __attribute__((ext_vector_type(2)));
typedef float        v8f __attribute__((ext_vector_type(8)));
typedef unsigned int v4u __attribute__((ext_vector_type(4)));
typedef int          v4i __attribute__((ext_vector_type(4)));
typedef int          v8i __attribute__((ext_vector_type(8)));

#define TILE_E      512        // edges per LDS tile (per buffer: 512*12B = 6 KB)
#define SCAT_BLOCKS 1024

// ---------------------------------------------------------------------------
// TDM: 1-D dword-tile DMA global -> LDS. Hand-packed D# (group0 + group1),
// trailing groups zero (<=2D tensor). amdgpu-toolchain 6-arg builtin form:
//   (uint32x4 g0, int32x8 g1, int32x4, int32x4, int32x8, i32 cpol)
// group0: [1:0]=count=1, [63:32]=lds_addr, [120:64]=global_addr, [127:126]=type=2
// group1: [17:16]=data_size=2 (4B), [79:48]=tensor_dim0, [127:112]=tile_dim0,
//         [207:160]=tensor_dim0_stride; workgroup_mask=0 (not in cluster).
// ---------------------------------------------------------------------------
__device__ __forceinline__ void tdm_load_dw_1d(const void* gptr,
                                               unsigned lds_byte_addr,
                                               unsigned nelem) {
  const unsigned long long ga = (unsigned long long)(uintptr_t)gptr;
  v4u g0;
  g0.x = 1u;                                            // count=1, user mode
  g0.y = lds_byte_addr;                                 // LDS dest (bytes)
  g0.z = (unsigned)(ga & 0xFFFFFFFFull);                // global_addr[31:0]
  g0.w = (unsigned)((ga >> 32) & 0x01FFFFFFull)         // global_addr[56:32]
       | (2u << 30);                                    // type = 2 ("image")
  v8i g1;
  g1[0] = (int)0x00020000u;                             // data_size = 4 bytes
  g1[1] = (int)((nelem & 0xFFFFu) << 16);               // tensor_dim0[15:0]
  g1[2] = (int)((nelem >> 16) & 0xFFFFu);               // tensor_dim0[31:16]
  g1[3] = (int)((nelem & 0xFFFFu) << 16);               // tile_dim0 (<=65535)
  g1[4] = 0;                                            // tile_dim1/2 unused
  g1[5] = (int)nelem;                                   // tensor_dim0_stride lo
  g1[6] = 0;
  g1[7] = 0;
  v4i z4 = {0, 0, 0, 0};
  v8i z8 = {0, 0, 0, 0, 0, 0, 0, 0};
  __builtin_amdgcn_tensor_load_to_lds(g0, g1, z4, z4, z8, 0);
}

// ---------------------------------------------------------------------------
// Init: emb = concat(user, item); acc = 0.25 * emb  (folds the /(L+1) mean)
// ---------------------------------------------------------------------------
__global__ void __launch_bounds__(256) lgcn_init(const float* __restrict__ ue,
                                                 const float* __restrict__ ie,
                                                 float* __restrict__ emb,
                                                 float* __restrict__ acc) {
  const size_t nU = (size_t)N_USERS * LDIM;
  for (size_t i = (size_t)blockIdx.x * blockDim.x + threadIdx.x; i < NELEM;
       i += (size_t)gridDim.x * blockDim.x) {
    const float v = (i < nU) ? ue[i] : ie[i - nU];
    emb[i] = v;
    acc[i] = 0.25f * v;
  }
}

__global__ void __launch_bounds__(256) lgcn_zero(float4* __restrict__ p, size_t n4) {
  const float4 z = make_float4(0.f, 0.f, 0.f, 0.f);
  for (size_t i = (size_t)blockIdx.x * blockDim.x + threadIdx.x; i < n4;
       i += (size_t)gridDim.x * blockDim.x)
    p[i] = z;
}

// ---------------------------------------------------------------------------
// SpMM scatter: nxt[row] += val * cur[col]. Edge tiles double-buffered in LDS
// via TDM; one wave per edge, lane l handles dims 2l..2l+1.
// cur (76.8 MB) and nxt (76.8 MB) are L2-resident on MI455X (192 MB L2).
// ---------------------------------------------------------------------------
__global__ void __launch_bounds__(256) lgcn_scatter(const float* __restrict__ vals,
                                                    const int*   __restrict__ rows,
                                                    const int*   __restrict__ cols,
                                                    const float* __restrict__ cur,
                                                    float*       __restrict__ nxt,
                                                    int nnz, int chunk) {
  __shared__ float sval[2][TILE_E];
  __shared__ int   srow[2][TILE_E];
  __shared__ int   scol[2][TILE_E];

  const int lane  = threadIdx.x & 31;
  const int wave  = threadIdx.x >> 5;
  const int start = blockIdx.x * chunk;
  const int end   = (start + chunk < nnz) ? (start + chunk) : nnz;
  if (start >= end) return;                       // uniform per block
  const int ntile = (end - start + TILE_E - 1) / TILE_E;

  if (wave == 0) {                                // prologue: tile 0 -> buf 0
    const int c0 = ((end - start) < TILE_E) ? (end - start) : TILE_E;
    tdm_load_dw_1d(vals + start, (unsigned)(uintptr_t)&sval[0][0], (unsigned)c0);
    tdm_load_dw_1d(rows + start, (unsigned)(uintptr_t)&srow[0][0], (unsigned)c0);
    tdm_load_dw_1d(cols + start, (unsigned)(uintptr_t)&scol[0][0], (unsigned)c0);
  }

  for (int t = 0; t < ntile; ++t) {
    const int buf = t & 1;
    if (wave == 0) {
      if (t + 1 < ntile) {                        // issue next tile, then wait
        const int ts1 = start + (t + 1) * TILE_E; // for the current 3 DMAs
        const int c1  = ((end - ts1) < TILE_E) ? (end - ts1) : TILE_E;
        const int nb  = buf ^ 1;
        tdm_load_dw_1d(vals + ts1, (unsigned)(uintptr_t)&sval[nb][0], (unsigned)c1);
        tdm_load_dw_1d(rows + ts1, (unsigned)(uintptr_t)&srow[nb][0], (unsigned)c1);
        tdm_load_dw_1d(cols + ts1, (unsigned)(uintptr_t)&scol[nb][0], (unsigned)c1);
        __builtin_amdgcn_s_wait_tensorcnt(3);     // oldest 3 (current tile) done
      } else {
        __builtin_amdgcn_s_wait_tensorcnt(0);
      }
    }
    __syncthreads();                              // publish LDS tile to all waves

    const int ts  = start + t * TILE_E;
    const int cnt = ((end - ts) < TILE_E) ? (end - ts) : TILE_E;
    for (int e = wave; e < cnt; e += 8) {
      const float v = sval[buf][e];               // LDS broadcast reads
      const int   c = scol[buf][e];
      const int   r = srow[buf][e];
      const float2 g = *(const float2*)(cur + (size_t)c * LDIM + lane * 2);
      float* dst = nxt + (size_t)r * LDIM + lane * 2;
      unsafeAtomicAdd(dst + 0, v * g.x);          // global_atomic_add_f32 (L2)
      unsafeAtomicAdd(dst + 1, v * g.y);
    }
    __syncthreads();                              // before buffer reuse
  }
}

// ---------------------------------------------------------------------------
// Layer-mean accumulate: acc += 0.25 * nxt, expressed per 16x16 tile as
// D = (0.25*I)(16x16) x E(16x16) + C via 4 chained v_wmma_f32_16x16x4_f32.
// Exact in f32 (entries 0 / 0.25). Documented wave32 layouts:
//  C/D: lane L, vgpr r  -> (M = r + 8*(L>>4), N = L&15)
//  A  : lane L, vgpr r  -> (M = L&15,        K = r + 2*(L>>4))
//  B  : lane L, vgpr r  -> (K = r + 2*(L>>4), N = L&15)   (mirror of A)
// ---------------------------------------------------------------------------
__global__ void __launch_bounds__(256) lgcn_accum(float* __restrict__ acc,
                                                  const float* __restrict__ nxt,
                                                  int ntiles) {
  const int lane = threadIdx.x & 31;
  const int half = lane >> 4;
  const int nidx = lane & 15;
  const int wid  = blockIdx.x * (blockDim.x >> 5) + (threadIdx.x >> 5);
  const int nw   = gridDim.x * (blockDim.x >> 5);

  for (int t = wid; t < ntiles; t += nw) {
    const int rb = t >> 2;                  // 16-row block (node rows)
    const int cb = t & 3;                   // 16-col block of the 64 dims
    const size_t base = (size_t)rb * 16 * LDIM + (size_t)cb * 16;

    v8f C;
#pragma unroll
    for (int r = 0; r < 8; ++r)
      C[r] = acc[base + (size_t)(r + 8 * half) * LDIM + nidx];

#if __has_builtin(__builtin_amdgcn_wmma_f32_16x16x4_f32)
#pragma unroll
    for (int kb = 0; kb < 4; ++kb) {
      v2f A, B;
#pragma unroll
      for (int r = 0; r < 2; ++r) {
        const int K = r + 2 * half;
        A[r] = (nidx == (4 * kb + K)) ? 0.25f : 0.0f;   // columns of 0.25*I
        B[r] = nxt[base + (size_t)(4 * kb + K) * LDIM + nidx];
      }
      C = __builtin_amdgcn_wmma_f32_16x16x4_f32(false, A, false, B,
                                                (short)0, C, false, false);
    }
#else // scalar fallback keeps the build green if the builtin name differs
#pragma unroll
    for (int r = 0; r < 8; ++r)
      C[r] += 0.25f * nxt[base + (size_t)(r + 8 * half) * LDIM + nidx];
#endif

#pragma unroll
    for (int r = 0; r < 8; ++r)
      acc[base + (size_t)(r + 8 * half) * LDIM + nidx] = C[r];
  }
}

// ---------------------------------------------------------------------------
extern "C" void kernel_launch(void* const* d_in, const int* in_sizes, int n_in,
                              void* d_out, int out_size, void* d_ws, size_t ws_size,
                              hipStream_t stream) {
  const float* ue   = (const float*)d_in[0];
  const float* ie   = (const float*)d_in[1];
  const float* vals = (const float*)d_in[2];
  const int*   rows = (const int*)d_in[3];
  const int*   cols = (const int*)d_in[4];

  float* acc  = (float*)d_out;          // running 0.25-scaled layer mean
  float* embA = (float*)d_ws;           // ping
  float* embB = embA + NELEM;           // pong

  lgcn_init<<<4096, 256, 0, stream>>>(ue, ie, embA, acc);

  const int chunk  = (NNZ_CNT + SCAT_BLOCKS - 1) / SCAT_BLOCKS;
  const int ntiles = (int)(NELEM / 256);          // 75000 16x16 tiles

  float* cur = embA;
  float* nxt = embB;
  for (int l = 0; l < NLAYERS; ++l) {
    lgcn_zero<<<4096, 256, 0, stream>>>((float4*)nxt, NELEM / 4);
    lgcn_scatter<<<SCAT_BLOCKS, 256, 0, stream>>>(vals, rows, cols, cur, nxt,
                                                  NNZ_CNT, chunk);
    lgcn_accum<<<2048, 256, 0, stream>>>(acc, nxt, ntiles);
    float* tmp = cur; cur = nxt; nxt = tmp;
  }
}